// SparseDecoder_27650999452105
// MI455X (gfx1250) — compile-verified
//
#include <hip/hip_runtime.h>

typedef __attribute__((ext_vector_type(2))) float v2f;
typedef __attribute__((ext_vector_type(8))) float v8f;

// One wave computes a full 64(batch) x 16(feature) output stripe: 4 stacked
// 16x16 WMMA accumulators share a single masked-weight (B) operand fetch, so
// each HBM weight/mask byte is loaded and mask-processed exactly once per
// K-split. Inner loop uses pointer-bump addressing so all loads are
// VGPR-base + constant immediate offset (no per-step 64-bit address math).
__global__ __launch_bounds__(128) void sparse_linear_wmma(
    const float* __restrict__ X,            // [64, K] activations
    const float* __restrict__ W,            // [N, K] weights
    const unsigned char* __restrict__ Mk,   // [N, K] boolean mask bytes
    const float* __restrict__ bias,         // [N]
    float* __restrict__ Y,                  // final [64,N] or partial [S,64,N]
    int K, int N, int Kchunk, int relu, int final_write)
{
  const int lane = threadIdx.x & 31;
  const int wave = threadIdx.x >> 5;       // 0..3: feature tile within block
  const int half = lane >> 4;              // K-pair select (WMMA f32 layout)
  const int l16  = lane & 15;

  const int n_tile = blockIdx.x * 4 + wave;
  const int col    = n_tile * 16 + l16;    // out feature (B operand N, D N)
  const int split  = blockIdx.y;
  const int klo    = split * Kchunk;

  // Per-lane streaming pointers with the lane's K-pair offset baked in.
  const int lk = klo + 2 * half;
  const float*         wp = W  + (size_t)col * K + lk;
  const unsigned char* mp = Mk + (size_t)col * K + lk;
  const float*         x0 = X + (size_t)(l16 +  0) * K + lk;
  const float*         x1 = X + (size_t)(l16 + 16) * K + lk;
  const float*         x2 = X + (size_t)(l16 + 32) * K + lk;
  const float*         x3 = X + (size_t)(l16 + 48) * K + lk;

  v8f acc0 = {}, acc1 = {}, acc2 = {}, acc3 = {};

  for (int it = Kchunk >> 4; it > 0; --it) {
#pragma unroll
    for (int u = 0; u < 4; ++u) {
      // B operand: Wm^T tile; mask bytes fused via exact 0/1 multiply
      // (v_cvt_f32_ubyte0/1 + v_mul_f32). Constant-offset loads.
      const unsigned int mw = *(const unsigned short*)(mp + 4 * u);
      v2f b;
      b.x = wp[4 * u + 0] * (float)(mw & 0xffu);
      b.y = wp[4 * u + 1] * (float)(mw >> 8);

      // A operands: same K pair from 4 batch tiles (L2-resident b64 loads).
      v2f a0, a1, a2, a3;
      a0.x = x0[4 * u]; a0.y = x0[4 * u + 1];
      a1.x = x1[4 * u]; a1.y = x1[4 * u + 1];
      a2.x = x2[4 * u]; a2.y = x2[4 * u + 1];
      a3.x = x3[4 * u]; a3.y = x3[4 * u + 1];

      acc0 = __builtin_amdgcn_wmma_f32_16x16x4_f32(false, a0, false, b, (short)0, acc0, false, false);
      acc1 = __builtin_amdgcn_wmma_f32_16x16x4_f32(false, a1, false, b, (short)0, acc1, false, false);
      acc2 = __builtin_amdgcn_wmma_f32_16x16x4_f32(false, a2, false, b, (short)0, acc2, false, false);
      acc3 = __builtin_amdgcn_wmma_f32_16x16x4_f32(false, a3, false, b, (short)0, acc3, false, false);
    }
    // One pointer bump per 16-K chunk.
    wp += 16; mp += 16;
    x0 += 16; x1 += 16; x2 += 16; x3 += 16;
  }

  // D layout: N = col, M(batch within tile) = v + 8*half.
  if (final_write) {
    const float bv = bias[col];
#pragma unroll
    for (int v = 0; v < 8; ++v) {
      const int bm = 8 * half + v;
      float r0 = acc0[v] + bv, r1 = acc1[v] + bv,
            r2 = acc2[v] + bv, r3 = acc3[v] + bv;
      if (relu) {
        r0 = fmaxf(r0, 0.0f); r1 = fmaxf(r1, 0.0f);
        r2 = fmaxf(r2, 0.0f); r3 = fmaxf(r3, 0.0f);
      }
      Y[(size_t)(bm +  0) * N + col] = r0;
      Y[(size_t)(bm + 16) * N + col] = r1;
      Y[(size_t)(bm + 32) * N + col] = r2;
      Y[(size_t)(bm + 48) * N + col] = r3;
    }
  } else {
    float* P = Y + (size_t)split * 64 * N;   // partial slab [64, N]
#pragma unroll
    for (int v = 0; v < 8; ++v) {
      const int bm = 8 * half + v;
      P[(size_t)(bm +  0) * N + col] = acc0[v];
      P[(size_t)(bm + 16) * N + col] = acc1[v];
      P[(size_t)(bm + 32) * N + col] = acc2[v];
      P[(size_t)(bm + 48) * N + col] = acc3[v];
    }
  }
}

// Deterministic K-split reduction: fixed-order sum + bias + optional ReLU.
__global__ __launch_bounds__(256) void reduce_bias_relu(
    const float* __restrict__ P, const float* __restrict__ bias,
    float* __restrict__ Y, int N, int S, int relu)
{
  const size_t i = (size_t)blockIdx.x * 256 + threadIdx.x;  // over 64*N
  const int n = (int)(i % (size_t)N);
  float s = 0.0f;
  for (int j = 0; j < S; ++j) s += P[(size_t)j * 64 * N + i];
  s += bias[n];
  if (relu) s = fmaxf(s, 0.0f);
  Y[i] = s;
}

extern "C" void kernel_launch(void* const* d_in, const int* in_sizes, int n_in,
                              void* d_out, int out_size, void* d_ws, size_t ws_size,
                              hipStream_t stream) {
  // Input order: x, W0, b0, mask0, W1, b1, mask1, W2, b2, mask2, W3, b3, mask3
  const float* x = (const float*)d_in[0];
  const float*         Wt[4];
  const float*         bs[4];
  const unsigned char* mk[4];
  for (int i = 0; i < 4; ++i) {
    Wt[i] = (const float*)d_in[1 + 3 * i];
    bs[i] = (const float*)d_in[2 + 3 * i];
    mk[i] = (const unsigned char*)d_in[3 + 3 * i];
  }

  const int B = 64;
  const int dims[5] = {512, 2048, 4096, 8192, 16384};

  // Workspace: activations first, then an optional K-split partial buffer.
  char* p = (char*)d_ws;
  float* act[3];
  act[0] = (float*)p;  p += (size_t)B * dims[1] * sizeof(float);
  act[1] = (float*)p;  p += (size_t)B * dims[2] * sizeof(float);
  act[2] = (float*)p;
  p += (size_t)B * dims[3] * sizeof(float);
  const size_t act_bytes = (size_t)(p - (char*)d_ws);
  float* part = (float*)p;
  const size_t part_avail = (ws_size > act_bytes) ? (ws_size - act_bytes) : 0;

  const float* cur = x;
  for (int i = 0; i < 4; ++i) {
    const int K = dims[i];
    const int N = dims[i + 1];
    const int relu = (i < 3) ? 1 : 0;
    float* Y = (i == 3) ? (float*)d_out : act[i];

    // Pick K-split so total waves ~>= 1024 (fill the WGPs on small layers),
    // but only if the partial slab fits in the workspace.
    const int base_waves = N / 16;
    int S = (base_waves >= 1024) ? 1
          : (base_waves >=  512) ? 2
          : (base_waves >=  256) ? 4 : 8;
    if (S > 1 && (size_t)S * B * N * sizeof(float) > part_avail) S = 1;

    dim3 grid(N / 64, S);   // 4 feature tiles per 128-thread block
    sparse_linear_wmma<<<grid, dim3(128), 0, stream>>>(
        cur, Wt[i], mk[i], bs[i], (S == 1) ? Y : part,
        K, N, K / S, relu, (S == 1) ? 1 : 0);

    if (S > 1) {
      reduce_bias_relu<<<dim3((B * N) / 256), dim3(256), 0, stream>>>(
          part, bs[i], Y, N, S, relu);
    }
    cur = Y;
  }
}